// FocalRegionLoss_39676907888504
// MI455X (gfx1250) — compile-verified
//
#include <hip/hip_runtime.h>

// FocalRegionLoss fused single-pass kernel for MI455X (gfx1250).
// Bandwidth-bound: 268 MB streamed once (~11.5 us @ 23.3 TB/s floor).
// Channel reduction done on the matrix unit: V_WMMA_F32_16X16X4_F32 Gram trick
// (A == B in registers => D += A*A^T, diagonal = per-pixel sum of squares, fp32).

#define BETA   2.0f
#define EPS2   (0.001f * 0.001f)
#define N_CLS  5
#define B_     16
#define C_     512
#define HW     4096           // 64*64
#define NPIX   (B_ * HW)      // 65536
#define NSEG   (B_ * N_CLS)   // 80

typedef __attribute__((ext_vector_type(2))) float v2f;
typedef __attribute__((ext_vector_type(8))) float v8f;

__global__ void frl_init(float* __restrict__ S, unsigned int* __restrict__ cnt) {
    int t = threadIdx.x;
    if (t < NSEG) { S[t] = 0.0f; cnt[t] = 0u; }
}

__global__ __launch_bounds__(256)
void frl_main(const float* __restrict__ rec,
              const float* __restrict__ aln,
              const int*   __restrict__ mask,
              float* __restrict__ S,
              unsigned int* __restrict__ cnt) {
    __shared__ float        sS[NSEG];
    __shared__ unsigned int sC[NSEG];
    if (threadIdx.x < NSEG) { sS[threadIdx.x] = 0.0f; sC[threadIdx.x] = 0u; }
    __syncthreads();

    const int lane = threadIdx.x & 31;
    const int wave = threadIdx.x >> 5;
    const int m    = lane & 15;   // pixel within 16-pixel tile (A-matrix row M)
    const int g    = lane >> 4;   // lane-half selects K pair {0,1} vs {2,3}
    const int wavesPerBlock = blockDim.x >> 5;
    const int nTiles = NPIX / 16; // 4096 tiles, one per wave

    for (int tile = blockIdx.x * wavesPerBlock + wave; tile < nTiles;
         tile += gridDim.x * wavesPerBlock) {
        const int b       = tile >> 8;          // 256 tiles per batch image
        const int pixbase = (tile & 255) << 4;  // 16 consecutive pixels, same b
        // A-matrix 16x4 f32 layout: lanes 0-15 hold K=k,k+1 ; lanes 16-31 hold K=k+2,k+3
        const int base = b * (C_ * HW) + (2 * g) * HW + pixbase + m;
        const float* rp = rec + base;
        const float* ap = aln + base;

        v8f acc = {0.f, 0.f, 0.f, 0.f, 0.f, 0.f, 0.f, 0.f};
        #pragma unroll 4
        for (int k = 0; k < C_; k += 4) {
            // Non-temporal: 268 MB streamed once, larger than the 192 MB L2.
            float r0 = __builtin_nontemporal_load(rp);
            float r1 = __builtin_nontemporal_load(rp + HW);
            float a0 = __builtin_nontemporal_load(ap);
            float a1 = __builtin_nontemporal_load(ap + HW);
            v2f d; d.x = r0 - a0; d.y = r1 - a1;
            // D += d * d^T : A and B share the same VGPR pair (layouts coincide).
            acc = __builtin_amdgcn_wmma_f32_16x16x4_f32(
                false, d, false, d, (short)0, acc, false, false);
            rp += 4 * HW; ap += 4 * HW;
        }

        // Diagonal extraction (C/D layout: VGPR j = row M=j (lanes 0-15) / M=j+8 (16-31)):
        // pixel p in [0,8):  acc[p]    at lane p
        // pixel p in [8,16): acc[p-8]  at lane p+16  (lanes 24..31)
        if (lane < 8 || lane >= 24) {
            const int j = (lane < 8) ? lane : (lane - 24);
            const int p = (lane < 8) ? lane : (lane - 16);
            float diag =
                (j == 0) ? acc[0] : (j == 1) ? acc[1] : (j == 2) ? acc[2] :
                (j == 3) ? acc[3] : (j == 4) ? acc[4] : (j == 5) ? acc[5] :
                (j == 6) ? acc[6] : acc[7];
            float loss = diag * (1.0f / (float)C_);      // mean over channels
            int cls = mask[b * HW + pixbase + p];
            int seg = b * N_CLS + cls;
            atomicAdd(&sS[seg], loss);
            atomicAdd(&sC[seg], 1u);
        }
    }
    __syncthreads();
    if (threadIdx.x < NSEG) {
        if (sC[threadIdx.x] != 0u) {
            atomicAdd(&S[threadIdx.x],   sS[threadIdx.x]);
            atomicAdd(&cnt[threadIdx.x], sC[threadIdx.x]);
        }
    }
}

__global__ void frl_final(const float* __restrict__ S,
                          const unsigned int* __restrict__ cnt,
                          float* __restrict__ out) {
    __shared__ float red[128];
    const int t = threadIdx.x;  // 128 threads
    float s = 0.0f, c = 0.0f;
    if (t < NSEG) { s = S[t]; c = (float)cnt[t]; }
    float avg = s / fmaxf(c, 1.0f);   // empty segments -> 0, never raises the max

    red[t] = avg;
    __syncthreads();
    for (int off = 64; off > 0; off >>= 1) {
        if (t < off) red[t] = fmaxf(red[t], red[t + off]);
        __syncthreads();
    }
    float wmax = red[0];
    __syncthreads();

    float w = (wmax > 0.0f) ? (avg / (wmax + EPS2)) : (avg + EPS2);
    w = fminf(fmaxf(w, 0.0f), 1.0f);
    // mean(loss*(w*BETA+1)) = (sum(S) + BETA * sum(w_seg * S_seg)) / NPIX
    red[t] = (t < NSEG) ? (s + BETA * w * s) : 0.0f;
    __syncthreads();
    for (int off = 64; off > 0; off >>= 1) {
        if (t < off) red[t] += red[t + off];
        __syncthreads();
    }
    if (t == 0) out[0] = red[0] * (1.0f / (float)NPIX);
}

extern "C" void kernel_launch(void* const* d_in, const int* in_sizes, int n_in,
                              void* d_out, int out_size, void* d_ws, size_t ws_size,
                              hipStream_t stream) {
    const float* rec  = (const float*)d_in[0];
    const float* aln  = (const float*)d_in[1];
    const int*   mask = (const int*)d_in[2];
    float*        S   = (float*)d_ws;
    unsigned int* cnt = (unsigned int*)((char*)d_ws + NSEG * sizeof(float));
    float* out = (float*)d_out;

    frl_init <<<1,   128, 0, stream>>>(S, cnt);
    frl_main <<<512, 256, 0, stream>>>(rec, aln, mask, S, cnt);  // 4096 tiles = 512 blk * 8 waves
    frl_final<<<1,   128, 0, stream>>>(S, cnt, out);
}